// HCDecoder_65008624993117
// MI455X (gfx1250) — compile-verified
//
#include <hip/hip_runtime.h>
#include <math.h>

typedef __attribute__((ext_vector_type(2))) float v2f;
typedef __attribute__((ext_vector_type(8))) float v8f;

#define TAU_INV 10.0f   // 1 / 0.1

// ---------------------------------------------------------------------------
// Projection GEMM via V_WMMA_F32_16X16X4_F32 (exact fp32 matrix path).
// P[b][m][n] = sum_k W[m][k] * X[b][k][n];  M = 64 fixed, C % 4 == 0, N % 32 == 0.
// 8 waves per block: wave -> (m_tile = wave&3, n_tile = wave>>2), 64x32 per block.
// ---------------------------------------------------------------------------
__global__ __launch_bounds__(256) void proj_wmma_kernel(
    const float* __restrict__ W,   // [64, C]
    const float* __restrict__ X,   // [B, C, N]
    float* __restrict__ P,         // [B, 64, N]
    int C, int N)
{
  const int b      = blockIdx.y;
  const int wave   = threadIdx.x >> 5;
  const int lane   = threadIdx.x & 31;
  const int m_base = (wave & 3) * 16;
  const int n_base = blockIdx.x * 32 + (wave >> 2) * 16;
  const int l15    = lane & 15;
  const int kh     = (lane >> 4) << 1;   // 0 for lanes 0-15, 2 for lanes 16-31

  const float* Wrow = W + (m_base + l15) * C;
  const float* Xb   = X + (size_t)b * C * N + n_base + l15;

  v8f acc = {};
  for (int k0 = 0; k0 < C; k0 += 4) {
    v2f a, bb;
    // A fragment (16x4): VGPR0 = K {0 | 2}, VGPR1 = K {1 | 3} per half-wave
    a.x = Wrow[k0 + kh];
    a.y = Wrow[k0 + kh + 1];
    // B fragment (4x16): symmetric K split, lane = pixel column (coalesced)
    const float* xp = Xb + (size_t)(k0 + kh) * N;
    bb.x = xp[0];
    bb.y = xp[(size_t)N];
    acc = __builtin_amdgcn_wmma_f32_16x16x4_f32(
        /*neg_a=*/false, a, /*neg_b=*/false, bb,
        /*c_mod=*/(short)0, acc, /*reuse_a=*/false, /*reuse_b=*/false);
  }
  // C/D layout: VGPR r -> row m_base + r (lanes 0-15) / m_base + 8 + r (lanes 16-31)
  float* Pp = P + (size_t)b * 64 * N + (size_t)(m_base + ((lane >> 4) << 3)) * N
                + n_base + l15;
#pragma unroll
  for (int r = 0; r < 8; ++r)
    Pp[(size_t)r * N] = acc[r];
}

// ---------------------------------------------------------------------------
// Fused cluster (sim + softmax) and upsample for one pyramid level.
// Block = 16x16 fine pixels -> 8x8 coarse cells (+1 halo = 10x10 tile in LDS).
// a_out[b,p,y,x] = softmax_p( <f[:,y,x], d[:, y/2+di-1, x/2+dj-1]> / TAU ),  p = di*3+dj
// o_out[b,ch,y,x] = sum_p a * co[b,ch, y/2+di-1, x/2+dj-1]     (zero-pad OOB)
// ---------------------------------------------------------------------------
template <int CC>
__global__ __launch_bounds__(256) void cluster_upsample_kernel(
    const float* __restrict__ f,    // [B,64,h,w]    fine projected features
    const float* __restrict__ d,    // [B,64,h/2,w/2] coarse projected features
    const float* __restrict__ co,   // [B,CC,h/2,w/2] coarse output
    float* __restrict__ a_out,      // [B,9,h,w]
    float* __restrict__ o_out,      // [B,CC,h,w]
    int h, int w)
{
  const int hc = h >> 1, wc = w >> 1;
  const int b  = blockIdx.z;
  const int tx = threadIdx.x, ty = threadIdx.y;
  const int tid = ty * 16 + tx;
  const int x = blockIdx.x * 16 + tx;
  const int y = blockIdx.y * 16 + ty;
  const int ci0 = blockIdx.y * 8 - 1;   // global coarse row of LDS tile row 0
  const int cj0 = blockIdx.x * 8 - 1;

  __shared__ float dtile[64 * 100];     // 25.6 KB
  __shared__ float cotile[CC * 100];    // 7.6 KB for CC=19

  const float* db = d + (size_t)b * 64 * hc * wc;
  for (int idx = tid; idx < 64 * 100; idx += 256) {
    int c = idx / 100, rem = idx - c * 100;
    int gi = ci0 + rem / 10, gj = cj0 + rem % 10;
    float v = 0.f;
    if (gi >= 0 && gi < hc && gj >= 0 && gj < wc)
      v = db[(size_t)c * hc * wc + gi * wc + gj];
    dtile[idx] = v;
  }
  const float* cob = co + (size_t)b * CC * hc * wc;
  for (int idx = tid; idx < CC * 100; idx += 256) {
    int c = idx / 100, rem = idx - c * 100;
    int gi = ci0 + rem / 10, gj = cj0 + rem % 10;
    float v = 0.f;
    if (gi >= 0 && gi < hc && gj >= 0 && gj < wc)
      v = cob[(size_t)c * hc * wc + gi * wc + gj];
    cotile[idx] = v;
  }
  __syncthreads();

  const int li = (y >> 1) - ci0;  // 1..8
  const int lj = (x >> 1) - cj0;

  float sim[9];
#pragma unroll
  for (int p = 0; p < 9; ++p) sim[p] = 0.f;

  const float* fb = f + (size_t)b * 64 * h * w + (size_t)y * w + x;
  for (int c = 0; c < 64; ++c) {
    float fv = fb[(size_t)c * h * w];
    const float* dl = &dtile[c * 100 + (li - 1) * 10 + (lj - 1)];
#pragma unroll
    for (int di = 0; di < 3; ++di)
#pragma unroll
      for (int dj = 0; dj < 3; ++dj)
        sim[di * 3 + dj] = fmaf(fv, dl[di * 10 + dj], sim[di * 3 + dj]);
  }

  // softmax(sim / TAU)
  float m = sim[0];
#pragma unroll
  for (int p = 1; p < 9; ++p) m = fmaxf(m, sim[p]);
  float ssum = 0.f;
#pragma unroll
  for (int p = 0; p < 9; ++p) {
    sim[p] = __expf((sim[p] - m) * TAU_INV);
    ssum += sim[p];
  }
  float inv = 1.0f / ssum;
  float* ab = a_out + (size_t)b * 9 * h * w + (size_t)y * w + x;
#pragma unroll
  for (int p = 0; p < 9; ++p) {
    sim[p] *= inv;
    ab[(size_t)p * h * w] = sim[p];
  }

  // upsample: weighted 3x3 gather of coarse output
  float* ob = o_out + (size_t)b * CC * h * w + (size_t)y * w + x;
  const float* col = &cotile[(li - 1) * 10 + (lj - 1)];
#pragma unroll
  for (int ch = 0; ch < CC; ++ch) {
    float o = 0.f;
    const float* cl = col + ch * 100;
#pragma unroll
    for (int di = 0; di < 3; ++di)
#pragma unroll
      for (int dj = 0; dj < 3; ++dj)
        o = fmaf(sim[di * 3 + dj], cl[di * 10 + dj], o);
    ob[(size_t)ch * h * w] = o;
  }
}

// ---------------------------------------------------------------------------
// Bilinear x4 resize, half-pixel centers, edge clamp (align_corners=False).
// ---------------------------------------------------------------------------
__global__ __launch_bounds__(256) void bilinear4x_kernel(
    const float* __restrict__ src,  // [BC, Hs, Ws]
    float* __restrict__ dst,        // [BC, Hd, Wd]
    int Hs, int Ws, int Hd, int Wd)
{
  const int X  = blockIdx.x * 16 + threadIdx.x;
  const int Y  = blockIdx.y * 16 + threadIdx.y;
  const int bc = blockIdx.z;
  if (X >= Wd || Y >= Hd) return;

  float sx = (X + 0.5f) * ((float)Ws / (float)Wd) - 0.5f;
  float sy = (Y + 0.5f) * ((float)Hs / (float)Hd) - 0.5f;
  float fxf = floorf(sx), fyf = floorf(sy);
  float fx = sx - fxf, fy = sy - fyf;
  int x0 = (int)fxf, y0 = (int)fyf;
  int x1 = min(x0 + 1, Ws - 1), y1 = min(y0 + 1, Hs - 1);
  x0 = max(x0, 0);
  y0 = max(y0, 0);

  const float* s = src + (size_t)bc * Hs * Ws;
  float v00 = s[(size_t)y0 * Ws + x0], v01 = s[(size_t)y0 * Ws + x1];
  float v10 = s[(size_t)y1 * Ws + x0], v11 = s[(size_t)y1 * Ws + x1];
  float v0 = v00 + (v01 - v00) * fx;
  float v1 = v10 + (v11 - v10) * fx;
  dst[(size_t)bc * Hd * Wd + (size_t)Y * Wd + X] = v0 + (v1 - v0) * fy;
}

// ---------------------------------------------------------------------------
extern "C" void kernel_launch(void* const* d_in, const int* in_sizes, int n_in,
                              void* d_out, int out_size, void* d_ws, size_t ws_size,
                              hipStream_t stream)
{
  const float* x_4   = (const float*)d_in[0];  // [4, 64,256,256]
  const float* s_8   = (const float*)d_in[1];  // [4,128,128,128]
  const float* x_8   = (const float*)d_in[2];  // [4,128,128,128]
  const float* s_16  = (const float*)d_in[3];  // [4,256, 64, 64]
  // d_in[4] = x (only H,W used: 1024x1024)
  const float* outp  = (const float*)d_in[5];  // [4, 19, 64, 64]
  const float* W_x4  = (const float*)d_in[6];  // [64, 64]
  const float* W_s8  = (const float*)d_in[7];  // [64,128]
  const float* W_x8  = (const float*)d_in[8];  // [64,128]
  const float* W_s16 = (const float*)d_in[9];  // [64,256]

  const int B = 4;

  // workspace layout (floats)
  float* ws   = (float*)d_ws;
  float* f8   = ws;                              // 4*64*16384
  float* d16  = f8   + (size_t)B * 64 * 16384;   // 4*64*4096
  float* f4   = d16  + (size_t)B * 64 * 4096;    // 4*64*65536
  float* d8   = f4   + (size_t)B * 64 * 65536;   // 4*64*16384
  float* o128 = d8   + (size_t)B * 64 * 16384;   // 4*19*16384
  float* o256 = o128 + (size_t)B * 19 * 16384;   // 4*19*65536

  // output layout: (a8, a4, out) concatenated
  float* a8   = (float*)d_out;                   // 4*9*128*128
  float* a4   = a8 + (size_t)B * 9 * 128 * 128;  // 4*9*256*256
  float* ofin = a4 + (size_t)B * 9 * 256 * 256;  // 4*19*1024*1024

  // --- 1x1-conv projections (WMMA f32 16x16x4) ---
  proj_wmma_kernel<<<dim3(16384 / 32, B), 256, 0, stream>>>(W_x8,  x_8,  f8,  128, 16384);
  proj_wmma_kernel<<<dim3(4096  / 32, B), 256, 0, stream>>>(W_s16, s_16, d16, 256, 4096);
  proj_wmma_kernel<<<dim3(65536 / 32, B), 256, 0, stream>>>(W_x4,  x_4,  f4,  64,  65536);
  proj_wmma_kernel<<<dim3(16384 / 32, B), 256, 0, stream>>>(W_s8,  s_8,  d8,  128, 16384);

  // --- level s=8: cluster(x_8 vs s_16) + upsample 64->128 ---
  cluster_upsample_kernel<19><<<dim3(128 / 16, 128 / 16, B), dim3(16, 16), 0, stream>>>(
      f8, d16, outp, a8, o128, 128, 128);

  // --- level s=4: cluster(x_4 vs s_8) + upsample 128->256 ---
  cluster_upsample_kernel<19><<<dim3(256 / 16, 256 / 16, B), dim3(16, 16), 0, stream>>>(
      f4, d8, o128, a4, o256, 256, 256);

  // --- bilinear x4 to 1024x1024 ---
  bilinear4x_kernel<<<dim3(1024 / 16, 1024 / 16, B * 19), dim3(16, 16), 0, stream>>>(
      o256, ofin, 256, 256, 1024, 1024);
}